// BiLSTM_CRF_12421045420739
// MI455X (gfx1250) — compile-verified
//
#include <hip/hip_runtime.h>
#include <hip/hip_bf16.h>

typedef float v2f __attribute__((ext_vector_type(2)));
typedef float v8f __attribute__((ext_vector_type(8)));

#define CRF_B 4096
#define CRF_T 512
#define CRF_C 15   // real tags; padded to 16 for WMMA

#define LOG2E 1.44269504088896340736f
#define LN2   0.69314718055994530942f

// Branchless per-lane select (forces v_cndmask/v_bfi instead of EXEC-branch regions)
__device__ __forceinline__ float sel_f32(bool c, float x, float y) {
  const unsigned m = c ? 0xffffffffu : 0u;
  return __uint_as_float((__float_as_uint(x) & m) | (__float_as_uint(y) & ~m));
}

// One wave (32 lanes) advances 16 sequences through the CRF forward scan.
//   D/C layout (alpha storage):   lane l, reg r  ->  state s = r + (l>=16?8:0), batch = l&15
//   A operand (exp(T^T), static): lane l, chunk c, comp v -> A[j=l&15][i=4c+v+(l>=16?2:0)]
//   B operand (p = exp(alpha-m)): lane l, chunk c, comp v -> B[i=4c+v+(l>=16?2:0)][b=l&15]
__global__ __launch_bounds__(32) void crf_fwd_kernel(
    const float* __restrict__ logits,     // [4096,512,15]
    const int*   __restrict__ y_true,     // [4096,512]
    const float* __restrict__ trans,      // [15,15]
    float*       __restrict__ nll_out)    // [4096]
{
  __shared__ float pshare[256];           // 16 states x 16 batches staging
  const int  lane  = threadIdx.x;         // 0..31 (wave32)
  const int  bb    = lane & 15;
  const bool up    = lane >= 16;
  const int  b     = blockIdx.x * 16 + bb;
  const int  khalf = up ? 2 : 0;
  const int  shalf = up ? 8 : 0;

  // ---- A operand: A[j][i] = exp(trans[i][j]), zero-padded to 16x16 (built once) ----
  v2f amat[4];
  #pragma unroll
  for (int c = 0; c < 4; ++c) {
    #pragma unroll
    for (int v = 0; v < 2; ++v) {
      const int i = 4 * c + v + khalf;    // K (prev state)
      const int j = bb;                   // M (next state)
      const bool ok = (i < CRF_C) && (j < CRF_C);
      const float tv = trans[(ok ? i : 0) * CRF_C + (ok ? j : 0)];
      const float val = sel_f32(ok, __builtin_amdgcn_exp2f(tv * LOG2E), 0.0f);
      if (v == 0) amat[c].x = val; else amat[c].y = val;
    }
  }

  // ---- alpha init: alpha0 = logits[:,0,:] (unconditional, per reference); pad state = -1e30 ----
  const float* row0 = logits + (size_t)b * CRF_T * CRF_C;
  float a[8];
  #pragma unroll
  for (int r = 0; r < 8; ++r) {
    const int s = r + shalf;
    a[r] = (s < CRF_C) ? row0[s] : -1.0e30f;
  }

  // ---- gold-path score state at t=0 (duplicated across both lane halves; harmless) ----
  const int  y0    = y_true[(size_t)b * CRF_T];
  bool       mprev = (y0 != 15);
  int        yprev = y0;
  float      em    = row0[mprev ? y0 : 0] * (mprev ? 1.0f : 0.0f);
  float      tr    = 0.0f;

  for (int t = 1; t < CRF_T; ++t) {
    const float* ep = logits + ((size_t)b * CRF_T + t) * CRF_C;
    if (t + 1 < CRF_T) __builtin_prefetch(ep + CRF_C, 0, 1);   // global_prefetch next row

    // m[b] = max_i alpha[b][i]  (8 in-lane + one cross-half shuffle)
    float m = a[0];
    #pragma unroll
    for (int r = 1; r < 8; ++r) m = fmaxf(m, a[r]);
    m = fmaxf(m, __shfl_xor(m, 16, 32));

    // p[i][b] = exp2((alpha-m)*log2e), staged D-layout -> B-layout through LDS
    #pragma unroll
    for (int r = 0; r < 8; ++r)
      pshare[(r + shalf) * 16 + bb] = __builtin_amdgcn_exp2f((a[r] - m) * LOG2E);
    __syncthreads();

    v2f bmat[4];
    #pragma unroll
    for (int c = 0; c < 4; ++c) {
      const int i0 = 4 * c + khalf;
      bmat[c].x = pshare[i0 * 16 + bb];
      bmat[c].y = pshare[(i0 + 1) * 16 + bb];
    }
    __syncthreads();

    // D[j][b] = sum_i exp(T[i][j]) * p[i][b] : four chained K=4 f32 WMMAs (K total = 16)
    v8f acc = {0.f, 0.f, 0.f, 0.f, 0.f, 0.f, 0.f, 0.f};
    #pragma unroll
    for (int c = 0; c < 4; ++c)
      acc = __builtin_amdgcn_wmma_f32_16x16x4_f32(false, amat[c], false, bmat[c],
                                                  (short)0, acc, false, false);

    const int  ybt   = y_true[(size_t)b * CRF_T + t];
    const bool maskt = (ybt != 15);

    // alpha'[s][b] = m + ln2*log2(D) + emit ; branchless masked update
    #pragma unroll
    for (int r = 0; r < 8; ++r) {
      const int s = r + shalf;
      const float e  = (s < CRF_C) ? ep[s] : 0.0f;
      const float lg = __builtin_amdgcn_logf(fmaxf(acc[r], 1.0e-30f));  // raw v_log_f32 (log2)
      const float na = m + LN2 * lg + e;
      a[r] = sel_f32(maskt, na, a[r]);
    }

    // gold-path emission + transition (safe indices -> unconditional loads, mask by multiply)
    const float mf = maskt ? 1.0f : 0.0f;
    em += ep[maskt ? ybt : 0] * mf;
    const bool tm  = maskt && mprev;
    const float tf = tm ? 1.0f : 0.0f;
    tr += trans[(tm ? yprev : 0) * CRF_C + (tm ? ybt : 0)] * tf;
    yprev = ybt;
    mprev = maskt;
  }

  // logZ[b] = logsumexp_s alpha[s][b]  (pad state contributes ~0)
  float m2 = a[0];
  #pragma unroll
  for (int r = 1; r < 8; ++r) m2 = fmaxf(m2, a[r]);
  m2 = fmaxf(m2, __shfl_xor(m2, 16, 32));
  float s2 = 0.0f;
  #pragma unroll
  for (int r = 0; r < 8; ++r) s2 += __builtin_amdgcn_exp2f((a[r] - m2) * LOG2E);
  s2 += __shfl_xor(s2, 16, 32);
  const float logZ = m2 + LN2 * __builtin_amdgcn_logf(s2);

  if (!up) {
    float nll = logZ - (em + tr);
    nll = fminf(fmaxf(nll, 0.0f), 1.0e6f);
    nll_out[b] = nll;
  }
}

// Deterministic tree reduction of the 4096 per-sequence NLLs -> mean scalar.
__global__ __launch_bounds__(256) void crf_reduce_kernel(
    const float* __restrict__ nll, float* __restrict__ out, int n)
{
  __shared__ float sm[256];
  const int tid = threadIdx.x;
  float s = 0.0f;
  for (int i = tid; i < n; i += 256) s += nll[i];
  sm[tid] = s;
  __syncthreads();
  #pragma unroll
  for (int k = 128; k > 0; k >>= 1) {
    if (tid < k) sm[tid] += sm[tid + k];
    __syncthreads();
  }
  if (tid == 0) out[0] = sm[0] / (float)n;
}

extern "C" void kernel_launch(void* const* d_in, const int* in_sizes, int n_in,
                              void* d_out, int out_size, void* d_ws, size_t ws_size,
                              hipStream_t stream) {
  const float* logits = (const float*)d_in[0];   // [4096,512,15] f32
  const int*   y_true = (const int*)  d_in[1];   // [4096,512]   i32
  const float* trans  = (const float*)d_in[2];   // [15,15]      f32
  float*       out    = (float*)d_out;           // scalar f32
  float*       nll_ws = (float*)d_ws;            // 4096 f32 scratch

  (void)in_sizes; (void)n_in; (void)out_size; (void)ws_size;

  crf_fwd_kernel<<<CRF_B / 16, 32, 0, stream>>>(logits, y_true, trans, nll_ws);
  crf_reduce_kernel<<<1, 256, 0, stream>>>(nll_ws, out, CRF_B);
}